// simpleGCN_15642270892451
// MI455X (gfx1250) — compile-verified
//
#include <hip/hip_runtime.h>
#include <hip/hip_bf16.h>
#include <math.h>

typedef __attribute__((ext_vector_type(2))) float v2f;
typedef __attribute__((ext_vector_type(8))) float v8f;

#define EPSBN 1e-5f

// ---------------------------------------------------------------------------
// Graph preprocessing
// ---------------------------------------------------------------------------
__global__ void gcn_fill1(float* __restrict__ deg, int N) {
    int i = blockIdx.x * blockDim.x + threadIdx.x;
    if (i < N) deg[i] = 1.0f;  // self-loop contribution
}

__global__ void gcn_deg_accum(const long long* __restrict__ dst, float* __restrict__ deg, int E) {
    int e = blockIdx.x * blockDim.x + threadIdx.x;
    if (e < E) atomicAdd(&deg[(int)dst[e]], 1.0f);
}

__global__ void gcn_rsqrt_inplace(float* __restrict__ d, int N) {
    int i = blockIdx.x * blockDim.x + threadIdx.x;
    if (i < N) d[i] = rsqrtf(d[i]);   // deg >= 1 always (self loop)
}

__global__ void gcn_edge_norm(const long long* __restrict__ src, const long long* __restrict__ dst,
                              const float* __restrict__ dinv, float* __restrict__ nrm, int E) {
    int e = blockIdx.x * blockDim.x + threadIdx.x;
    if (e < E) nrm[e] = dinv[(int)src[e]] * dinv[(int)dst[e]];
}

__global__ void gcn_pad_w3(const float* __restrict__ W3, float* __restrict__ Wp) {
    int t = blockIdx.x * blockDim.x + threadIdx.x;   // 256*48
    if (t < 256 * 48) {
        int k = t / 48, c = t % 48;
        Wp[t] = (c < 40) ? W3[k * 40 + c] : 0.0f;
    }
}

// ---------------------------------------------------------------------------
// GEMM: D[N x C] = A[N x K] * B[K x C], fp32 via V_WMMA_F32_16X16X4_F32.
// One 16x16 output tile per wave; 8 waves/block share a 16xK A-panel in LDS.
// N multiple of 16 (50000 = 3125*16). C multiple of 16 (256 or 48).
// ---------------------------------------------------------------------------
__global__ __launch_bounds__(256)
void gcn_gemm_wmma(const float* __restrict__ A, const float* __restrict__ B,
                   float* __restrict__ D, int K, int C) {
    __shared__ float As[16][260];  // padded stride: kills 16-way bank conflicts
    const int rowTile = blockIdx.x;
    const int lane = threadIdx.x & 31;
    const int wave = threadIdx.x >> 5;
    const int colTile = blockIdx.y * 8 + wave;

    const float* Ap = A + (size_t)rowTile * 16 * K;
    for (int idx = threadIdx.x; idx < 16 * K; idx += 256) {
        int r = idx / K, c = idx - r * K;
        As[r][c] = Ap[idx];
    }
    __syncthreads();

    if (colTile * 16 < C) {               // wave-uniform: EXEC all-ones inside
        v8f acc = {};
        const int r    = lane & 15;
        const int koff = (lane < 16) ? 0 : 2;
        const int col  = colTile * 16 + (lane & 15);
        for (int k = 0; k < K; k += 4) {
            v2f a, b;
            a.x = As[r][k + koff + 0];
            a.y = As[r][k + koff + 1];
            b.x = B[(size_t)(k + koff + 0) * C + col];
            b.y = B[(size_t)(k + koff + 1) * C + col];
            acc = __builtin_amdgcn_wmma_f32_16x16x4_f32(
                /*neg_a=*/false, a, /*neg_b=*/false, b,
                /*c_mod=*/(short)0, acc, /*reuse_a=*/false, /*reuse_b=*/false);
        }
        float* Dp = D + (size_t)rowTile * 16 * C + col;
        const int rbase = (lane < 16) ? 0 : 8;
#pragma unroll
        for (int j = 0; j < 8; ++j)
            Dp[(size_t)(rbase + j) * C] = acc[j];
    }
}

// ---------------------------------------------------------------------------
// out[i,c] = m[i,c]*dinv[i]^2 + b[c]  (self-loop message + bias; inits out)
// float4-vectorized streaming pass.
// ---------------------------------------------------------------------------
__global__ void gcn_selfloop_init4(const float4* __restrict__ m, const float* __restrict__ dinv,
                                   const float* __restrict__ bias, float4* __restrict__ out,
                                   long long total4, int c4, int blen) {
    long long t = (long long)blockIdx.x * blockDim.x + threadIdx.x;
    if (t >= total4) return;
    int i = (int)(t / c4);
    int q = (int)(t - (long long)i * c4) * 4;
    float di = dinv[i];
    float di2 = di * di;
    float4 v = m[t];
    v.x = v.x * di2 + ((q + 0 < blen) ? bias[q + 0] : 0.0f);
    v.y = v.y * di2 + ((q + 1 < blen) ? bias[q + 1] : 0.0f);
    v.z = v.z * di2 + ((q + 2 < blen) ? bias[q + 2] : 0.0f);
    v.w = v.w * di2 + ((q + 3 < blen) ? bias[q + 3] : 0.0f);
    out[t] = v;
}

// ---------------------------------------------------------------------------
// Edge scatter-add: out[dst] += m[src] * norm[e].
// One wave per edge; lanes stride over float4 channel chunks.
// No integer division; edge indices fetched once per wave (one cacheline).
// fp32 atomics resolve in L2 (feature matrix is L2-resident).
// ---------------------------------------------------------------------------
__global__ __launch_bounds__(256)
void gcn_edge_scatter(const float* __restrict__ m, const long long* __restrict__ src,
                      const long long* __restrict__ dst, const float* __restrict__ nrm,
                      float* __restrict__ out, int E, int C) {
    const int wave = threadIdx.x >> 5;
    const int lane = threadIdx.x & 31;
    const int e = blockIdx.x * 8 + wave;
    if (e >= E) return;
    const int s = (int)src[e];
    const int d = (int)dst[e];
    const float w = nrm[e];
    const float4* mrow = (const float4*)(m + (size_t)s * C);
    float*        orow = out + (size_t)d * C;
    const int c4 = C >> 2;
    for (int q = lane; q < c4; q += 32) {
        float4 v = mrow[q];
        float* o = orow + q * 4;
        atomicAdd(o + 0, v.x * w);
        atomicAdd(o + 1, v.y * w);
        atomicAdd(o + 2, v.z * w);
        atomicAdd(o + 3, v.w * w);
    }
}

// ---------------------------------------------------------------------------
// BatchNorm (training stats) + ReLU, fused apply. C fixed at 256 for BN layers.
// ---------------------------------------------------------------------------
__global__ void gcn_zero_stats(float* __restrict__ stats) {
    int t = blockIdx.x * blockDim.x + threadIdx.x;
    if (t < 512) stats[t] = 0.0f;
}

__global__ __launch_bounds__(256)
void gcn_bn_stats(const float* __restrict__ h, float* __restrict__ stats, int N) {
    const int c = threadIdx.x;                 // channel, C == 256
    const int r0 = blockIdx.x * 128;
    const int r1 = min(r0 + 128, N);
    float s = 0.0f, sq = 0.0f;
    for (int r = r0; r < r1; ++r) {
        float v = h[(size_t)r * 256 + c];
        s += v;
        sq += v * v;
    }
    atomicAdd(&stats[c], s);
    atomicAdd(&stats[256 + c], sq);
}

__global__ void gcn_bn_finalize(float* __restrict__ stats, int N) {
    int c = threadIdx.x;                       // <<<1,256>>>
    float invN = 1.0f / (float)N;
    float mu = stats[c] * invN;
    float var = stats[256 + c] * invN - mu * mu;
    stats[512 + c] = mu;
    stats[768 + c] = rsqrtf(var + EPSBN);
}

__global__ void gcn_bn_apply_relu4(float4* __restrict__ h, const float* __restrict__ stats,
                                   const float* __restrict__ g, const float* __restrict__ be,
                                   long long total4) {
    long long t = (long long)blockIdx.x * blockDim.x + threadIdx.x;
    if (t >= total4) return;
    int q = (int)(t & 63) * 4;                 // C == 256 -> 64 float4 per row
    float4 v = h[t];
    v.x = fmaxf((v.x - stats[512 + q + 0]) * stats[768 + q + 0] * g[q + 0] + be[q + 0], 0.0f);
    v.y = fmaxf((v.y - stats[512 + q + 1]) * stats[768 + q + 1] * g[q + 1] + be[q + 1], 0.0f);
    v.z = fmaxf((v.z - stats[512 + q + 2]) * stats[768 + q + 2] * g[q + 2] + be[q + 2], 0.0f);
    v.w = fmaxf((v.w - stats[512 + q + 3]) * stats[768 + q + 3] * g[q + 3] + be[q + 3], 0.0f);
    h[t] = v;
}

// ---------------------------------------------------------------------------
// Head: y = a(40) @ Wl(40x40) + bl, then log_softmax.  One thread per row.
// ---------------------------------------------------------------------------
__global__ void gcn_head(const float* __restrict__ agg, const float* __restrict__ Wl,
                         const float* __restrict__ bl, float* __restrict__ out, int N) {
    int i = blockIdx.x * blockDim.x + threadIdx.x;
    if (i >= N) return;
    const float* row = agg + (size_t)i * 48;   // padded stride, first 40 valid
    float a[40];
#pragma unroll
    for (int k = 0; k < 40; ++k) a[k] = row[k];
    float y[40];
    for (int o = 0; o < 40; ++o) {
        float s = bl[o];
        for (int k = 0; k < 40; ++k) s += a[k] * Wl[k * 40 + o];
        y[o] = s;
    }
    float mx = y[0];
    for (int o = 1; o < 40; ++o) mx = fmaxf(mx, y[o]);
    float se = 0.0f;
    for (int o = 0; o < 40; ++o) se += expf(y[o] - mx);
    float lse = mx + logf(se);
    float* op = out + (size_t)i * 40;
    for (int o = 0; o < 40; ++o) op[o] = y[o] - lse;
}

// ---------------------------------------------------------------------------
extern "C" void kernel_launch(void* const* d_in, const int* in_sizes, int n_in,
                              void* d_out, int out_size, void* d_ws, size_t ws_size,
                              hipStream_t stream) {
    const float*      x   = (const float*)d_in[0];
    const long long*  ei  = (const long long*)d_in[1];   // int64 in reference
    const float*      W1  = (const float*)d_in[2];
    const float*      b1  = (const float*)d_in[3];
    const float*      g1  = (const float*)d_in[4];
    const float*      be1 = (const float*)d_in[5];
    const float*      W2  = (const float*)d_in[6];
    const float*      b2  = (const float*)d_in[7];
    const float*      g2  = (const float*)d_in[8];
    const float*      be2 = (const float*)d_in[9];
    const float*      W3  = (const float*)d_in[10];
    const float*      b3  = (const float*)d_in[11];
    const float*      Wl  = (const float*)d_in[12];
    const float*      bl  = (const float*)d_in[13];

    const int N = in_sizes[0] / 128;   // 50000
    const int E = in_sizes[1] / 2;     // 800000
    const long long* src = ei;
    const long long* dst = ei + E;

    // Workspace layout (floats)
    float* ws    = (float*)d_ws;
    float* dinv  = ws;                       // N      (deg, then rsqrt in place)
    float* nrm   = dinv + N;                 // E
    float* stats = nrm + E;                  // 1024: sum|sumsq|mu|invstd
    float* wpad  = stats + 1024;             // 256*48
    float* mbuf  = wpad + 256 * 48;          // N*256  (GEMM output / messages)
    float* abuf  = mbuf + (size_t)N * 256;   // N*256  (aggregate / activations)

    const long long tot256_4 = (long long)N * 64;   // N*256/4
    const long long tot48_4  = (long long)N * 12;   // N*48/4
    const int gN    = (N + 255) / 256;
    const int gE    = (E + 255) / 256;
    const int g256v = (int)((tot256_4 + 255) / 256);
    const int g48v  = (int)((tot48_4 + 255) / 256);
    const int gEw   = (E + 7) / 8;                  // one wave per edge, 8 waves/block

    // --- preprocessing: degrees, dinv, edge norms, padded W3 ---
    gcn_fill1<<<gN, 256, 0, stream>>>(dinv, N);
    gcn_deg_accum<<<gE, 256, 0, stream>>>(dst, dinv, E);
    gcn_rsqrt_inplace<<<gN, 256, 0, stream>>>(dinv, N);
    gcn_edge_norm<<<gE, 256, 0, stream>>>(src, dst, dinv, nrm, E);
    gcn_pad_w3<<<48, 256, 0, stream>>>(W3, wpad);

    // --- layer 1: GCNConv(128->256) + BN + ReLU ---
    gcn_gemm_wmma<<<dim3(N / 16, 2), 256, 0, stream>>>(x, W1, mbuf, 128, 256);
    gcn_selfloop_init4<<<g256v, 256, 0, stream>>>((const float4*)mbuf, dinv, b1,
                                                  (float4*)abuf, tot256_4, 64, 256);
    gcn_edge_scatter<<<gEw, 256, 0, stream>>>(mbuf, src, dst, nrm, abuf, E, 256);
    gcn_zero_stats<<<2, 256, 0, stream>>>(stats);
    gcn_bn_stats<<<(N + 127) / 128, 256, 0, stream>>>(abuf, stats, N);
    gcn_bn_finalize<<<1, 256, 0, stream>>>(stats, N);
    gcn_bn_apply_relu4<<<g256v, 256, 0, stream>>>((float4*)abuf, stats, g1, be1, tot256_4);

    // --- layer 2: GCNConv(256->256) + BN + ReLU ---
    gcn_gemm_wmma<<<dim3(N / 16, 2), 256, 0, stream>>>(abuf, W2, mbuf, 256, 256);
    gcn_selfloop_init4<<<g256v, 256, 0, stream>>>((const float4*)mbuf, dinv, b2,
                                                  (float4*)abuf, tot256_4, 64, 256);
    gcn_edge_scatter<<<gEw, 256, 0, stream>>>(mbuf, src, dst, nrm, abuf, E, 256);
    gcn_zero_stats<<<2, 256, 0, stream>>>(stats);
    gcn_bn_stats<<<(N + 127) / 128, 256, 0, stream>>>(abuf, stats, N);
    gcn_bn_finalize<<<1, 256, 0, stream>>>(stats, N);
    gcn_bn_apply_relu4<<<g256v, 256, 0, stream>>>((float4*)abuf, stats, g2, be2, tot256_4);

    // --- layer 3: GCNConv(256->40), padded to 48 cols ---
    gcn_gemm_wmma<<<dim3(N / 16, 1), 256, 0, stream>>>(abuf, wpad, mbuf, 256, 48);
    gcn_selfloop_init4<<<g48v, 256, 0, stream>>>((const float4*)mbuf, dinv, b3,
                                                 (float4*)abuf, tot48_4, 12, 40);
    gcn_edge_scatter<<<gEw, 256, 0, stream>>>(mbuf, src, dst, nrm, abuf, E, 48);

    // --- head: Linear(40->40) + log_softmax ---
    gcn_head<<<(N + 127) / 128, 128, 0, stream>>>(abuf, Wl, bl, (float*)d_out, N);
}